// MambaBlock_73572789780827
// MI455X (gfx1250) — compile-verified
//
#include <hip/hip_runtime.h>
#include <cstdint>

// ---- problem constants (match reference) ----
#define D_MODEL  768
#define D_STATE  16
#define D_CONV   4
#define D_INNER  1536          // D_MODEL * 2
#define DT_RANK  48            // D_MODEL / 16
#define BATCH    2
#define SEQ_L    2048
#define M_ROWS   (BATCH * SEQ_L)         // 4096 tokens
#define XPROJ_N  (DT_RANK + 2 * D_STATE) // 80

typedef float v2f __attribute__((ext_vector_type(2)));
typedef float v8f __attribute__((ext_vector_type(8)));

__device__ __forceinline__ float sigmoid_f(float x) { return 1.0f / (1.0f + __expf(-x)); }
__device__ __forceinline__ float softplus_f(float x) { return (x > 20.0f) ? x : log1pf(__expf(x)); }

// ---------------------------------------------------------------------------
// FP32 WMMA GEMM:  C[M,N] = act( A[M,K] * W[N,K]^T + bias[N] )
//
// Block = 256 threads (8 waves), tile = 128 x (16*NT), K staged in chunks of
// KC through double-buffered LDS:
//   - coalesced global float4 loads -> regs (issued one chunk ahead)
//   - ds_store_b128 into LDS rows padded to STR=KC+4 floats
//     (STR%4==0 keeps 16B alignment; STR*row mod 64 is distinct for
//      row=0..15 -> fragment ds_load_b64 is bank-conflict free)
//   - per k-step: 1 A-frag + NT B-frag ds_load_b64 feeding NT WMMAs
// Fragment layouts per CDNA5 ISA 7.12.2 (32-bit A 16x4 / B 4x16 / C 16x16).
// ACT: 0 = none, 1 = softplus
// ---------------------------------------------------------------------------
template <int NT, int KC, int ACT>
__global__ __launch_bounds__(256) void gemm_wmma_f32(
    const float* __restrict__ A, int lda,
    const float* __restrict__ W, int ldw,
    const float* __restrict__ bias,
    float* __restrict__ C, int ldc,
    int K)
{
    constexpr int MB   = 128;
    constexpr int NB   = 16 * NT;
    constexpr int STR  = KC + 4;             // padded LDS row stride (floats)
    constexpr int KQ   = KC / 4;             // float4s per row per chunk
    constexpr int APT  = (MB * KQ) / 256;    // A float4s per thread
    constexpr int BTOT = NB * KQ;
    constexpr int BPT  = (BTOT + 255) / 256; // B float4s per thread (ceil)

    __shared__ float sA[2][MB * STR];
    __shared__ float sB[2][NB * STR];

    const int tid  = threadIdx.x;
    const int lane = tid & 31;
    const int wv   = tid >> 5;
    const int half = lane >> 4;
    const int l15  = lane & 15;

    const int m_blk = blockIdx.y * MB;
    const int n_blk = blockIdx.x * NB;

    float4 ra[APT];
    float4 rb[BPT];

    auto gload = [&](int kc) {
#pragma unroll
        for (int n = 0; n < APT; n++) {
            const int i = tid + n * 256;
            const int row = i / KQ, cq = i % KQ;
            ra[n] = *(const float4*)(A + (size_t)(m_blk + row) * lda + kc + cq * 4);
        }
#pragma unroll
        for (int n = 0; n < BPT; n++) {
            const int i = tid + n * 256;
            if (i < BTOT) {
                const int row = i / KQ, cq = i % KQ;
                rb[n] = *(const float4*)(W + (size_t)(n_blk + row) * ldw + kc + cq * 4);
            }
        }
    };
    auto lstore = [&](int buf) {
#pragma unroll
        for (int n = 0; n < APT; n++) {
            const int i = tid + n * 256;
            const int row = i / KQ, cq = i % KQ;
            *(float4*)&sA[buf][row * STR + cq * 4] = ra[n];
        }
#pragma unroll
        for (int n = 0; n < BPT; n++) {
            const int i = tid + n * 256;
            if (i < BTOT) {
                const int row = i / KQ, cq = i % KQ;
                *(float4*)&sB[buf][row * STR + cq * 4] = rb[n];
            }
        }
    };

    v8f acc[NT] = {};
    const int NC = K / KC;

    gload(0);
    lstore(0);
    __syncthreads();

    for (int c = 0; c < NC; c++) {
        if (c + 1 < NC) gload((c + 1) * KC);   // overlap HBM latency with WMMAs

        const float* As = &sA[c & 1][(wv * 16 + l15) * STR + 2 * half];
        const float* Bs = &sB[c & 1][l15 * STR + 2 * half];
#pragma unroll
        for (int ks = 0; ks < KQ; ks++) {
            const v2f av = *(const v2f*)(As + ks * 4);
#pragma unroll
            for (int j = 0; j < NT; j++) {
                const v2f bv = *(const v2f*)(Bs + j * 16 * STR + ks * 4);
                acc[j] = __builtin_amdgcn_wmma_f32_16x16x4_f32(
                    false, av, false, bv, (short)0, acc[j], false, false);
            }
        }
        __syncthreads();
        if (c + 1 < NC) {
            lstore((c + 1) & 1);
            __syncthreads();
        }
    }

    const int m0 = m_blk + wv * 16;
#pragma unroll
    for (int j = 0; j < NT; j++) {
        const int col = n_blk + j * 16 + l15;
        const float bval = bias ? bias[col] : 0.0f;
#pragma unroll
        for (int r = 0; r < 8; r++) {
            const int row = m0 + r + 8 * half;
            float v = acc[j][r] + bval;
            if (ACT == 1) v = softplus_f(v);
            C[(size_t)row * ldc + col] = v;
        }
    }
}

// ---------------------------------------------------------------------------
// Depthwise causal conv (k=4) + bias + SiLU over x_ssm (first half of xz).
// ---------------------------------------------------------------------------
__global__ __launch_bounds__(256) void conv_silu_kernel(
    const float* __restrict__ xz,
    const float* __restrict__ cw,   // (D_INNER, 1, 4)
    const float* __restrict__ cb,   // (D_INNER,)
    float* __restrict__ xc)
{
    const size_t idx = (size_t)blockIdx.x * blockDim.x + threadIdx.x;
    const int    d   = (int)(idx % D_INNER);
    const size_t bl  = idx / D_INNER;          // b*L + l
    const int    l   = (int)(bl % SEQ_L);

    float acc = cb[d];
#pragma unroll
    for (int j = 0; j < D_CONV; j++) {
        const int t = l - (D_CONV - 1) + j;
        if (t >= 0)
            acc = fmaf(cw[d * D_CONV + j],
                       xz[(bl - (size_t)(D_CONV - 1 - j)) * (2 * D_INNER) + d], acc);
    }
    xc[idx] = acc * sigmoid_f(acc);            // SiLU
}

// ---------------------------------------------------------------------------
// Selective-scan: one thread per (b, d), 16-state h in VGPRs.
// B_t/C_t double-buffered in LDS -> one barrier per timestep.
// Fuses  y = (scan + u*D) * silu(z).
// ---------------------------------------------------------------------------
__global__ __launch_bounds__(256) void ssm_scan_kernel(
    const float* __restrict__ dbc,    // [b, l, 80] : [dt(48) | B(16) | C(16)]
    const float* __restrict__ delta,  // [b, l, 1536]
    const float* __restrict__ u,      // [b, l, 1536]  (xc)
    const float* __restrict__ xz,     // [b, l, 3072]  (z = second half)
    const float* __restrict__ A_log,  // [1536, 16]
    const float* __restrict__ Dvec,   // [1536]
    float* __restrict__ ymod)         // [b, l, 1536]
{
    const int tid = threadIdx.x;
    const int d   = blockIdx.x * 256 + tid;
    const int b   = blockIdx.y;

    __shared__ float sB[2][D_STATE];
    __shared__ float sC[2][D_STATE];

    float a[D_STATE], h[D_STATE];
#pragma unroll
    for (int s = 0; s < D_STATE; s++) {
        a[s] = -__expf(A_log[d * D_STATE + s]);
        h[s] = 0.0f;
    }
    const float Dd = Dvec[d];

    const float* dbcb = dbc   + (size_t)b * SEQ_L * XPROJ_N + DT_RANK;
    const float* delb = delta + (size_t)b * SEQ_L * D_INNER + d;
    const float* ub   = u     + (size_t)b * SEQ_L * D_INNER + d;
    const float* zb   = xz    + (size_t)b * SEQ_L * (2 * D_INNER) + D_INNER + d;
    float*       yb   = ymod  + (size_t)b * SEQ_L * D_INNER + d;

    // prologue: B_0 / C_0 into buffer 0
    if (tid < 2 * D_STATE) {
        const float v = dbcb[tid];
        if (tid < D_STATE) sB[0][tid] = v;
        else               sC[0][tid - D_STATE] = v;
    }
    __syncthreads();

    for (int t = 0; t < SEQ_L; t++) {
        const int cur = t & 1, nxt = cur ^ 1;
        if (t + 1 < SEQ_L && tid < 2 * D_STATE) {   // stage t+1 into other buffer
            const float v = dbcb[(size_t)(t + 1) * XPROJ_N + tid];
            if (tid < D_STATE) sB[nxt][tid] = v;
            else               sC[nxt][tid - D_STATE] = v;
        }

        const float dt = delb[(size_t)t * D_INNER];
        const float ut = ub[(size_t)t * D_INNER];
        const float zt = zb[(size_t)t * (2 * D_INNER)];
        if (t + 1 < SEQ_L) {   // hide next-step latency: global_prefetch
            __builtin_prefetch(delb + (size_t)(t + 1) * D_INNER, 0, 3);
            __builtin_prefetch(ub   + (size_t)(t + 1) * D_INNER, 0, 3);
            __builtin_prefetch(zb   + (size_t)(t + 1) * (2 * D_INNER), 0, 3);
        }

        const float dtu = dt * ut;
        float y = 0.0f;
#pragma unroll
        for (int s = 0; s < D_STATE; s++) {
            const float dA = __expf(dt * a[s]);
            h[s] = fmaf(dA, h[s], dtu * sB[cur][s]);
            y    = fmaf(h[s], sC[cur][s], y);
        }
        y = fmaf(ut, Dd, y);
        y *= zt * sigmoid_f(zt);               // * silu(z)
        yb[(size_t)t * D_INNER] = y;
        __syncthreads();
    }
}

// ---------------------------------------------------------------------------
extern "C" void kernel_launch(void* const* d_in, const int* in_sizes, int n_in,
                              void* d_out, int out_size, void* d_ws, size_t ws_size,
                              hipStream_t stream)
{
    (void)in_sizes; (void)n_in; (void)out_size; (void)ws_size;

    const float* x      = (const float*)d_in[0];
    const float* W_in   = (const float*)d_in[1];
    const float* conv_w = (const float*)d_in[2];
    const float* conv_b = (const float*)d_in[3];
    const float* W_x    = (const float*)d_in[4];
    const float* W_dt   = (const float*)d_in[5];
    const float* b_dt   = (const float*)d_in[6];
    const float* A_log  = (const float*)d_in[7];
    const float* Dv     = (const float*)d_in[8];
    const float* W_out  = (const float*)d_in[9];
    float* out = (float*)d_out;

    float* ws = (float*)d_ws;
    size_t o = 0;
    float* xz    = ws + o; o += (size_t)M_ROWS * 2 * D_INNER;
    float* xc    = ws + o; o += (size_t)M_ROWS * D_INNER;
    float* dbc   = ws + o; o += (size_t)M_ROWS * XPROJ_N;
    float* delta = ws + o; o += (size_t)M_ROWS * D_INNER;
    float* ymod  = ws + o; o += (size_t)M_ROWS * D_INNER;   // ~127 MB total

    // 1) xz = x @ W_in^T          (M=4096, K=768, N=3072)
    gemm_wmma_f32<4, 32, 0><<<dim3((2 * D_INNER) / 64, M_ROWS / 128), 256, 0, stream>>>(
        x, D_MODEL, W_in, D_MODEL, nullptr, xz, 2 * D_INNER, D_MODEL);

    // 2) xc = silu(depthwise_conv(x_ssm) + conv_b)
    conv_silu_kernel<<<(M_ROWS * D_INNER) / 256, 256, 0, stream>>>(xz, conv_w, conv_b, xc);

    // 3) dbc = xc @ W_x^T         (M=4096, K=1536, N=80)
    gemm_wmma_f32<5, 32, 0><<<dim3(1, M_ROWS / 128), 256, 0, stream>>>(
        xc, D_INNER, W_x, D_INNER, nullptr, dbc, XPROJ_N, D_INNER);

    // 4) delta = softplus(dbc[:, :48] @ W_dt^T + b_dt)   (K=48, N=1536)
    gemm_wmma_f32<4, 16, 1><<<dim3(D_INNER / 64, M_ROWS / 128), 256, 0, stream>>>(
        dbc, XPROJ_N, W_dt, DT_RANK, b_dt, delta, D_INNER, DT_RANK);

    // 5) selective scan -> ymod = (scan + u*D) * silu(z)
    ssm_scan_kernel<<<dim3(D_INNER / 256, BATCH), 256, 0, stream>>>(
        dbc, delta, xc, xz, A_log, Dv, ymod);

    // 6) out = ymod @ W_out^T     (M=4096, K=1536, N=768)
    gemm_wmma_f32<4, 32, 0><<<dim3(D_MODEL / 64, M_ROWS / 128), 256, 0, stream>>>(
        ymod, D_INNER, W_out, D_INNER, nullptr, out, D_MODEL, D_INNER);
}